// RadialDescriptor_61847529062468
// MI455X (gfx1250) — compile-verified
//
#include <hip/hip_runtime.h>
#include <math.h>

#define N_ATOMS   50000
#define NN        32
#define N_TYPES   4
#define N_DESC    8
#define K_MAX     8
#define R_C       5.0f
#define PI_F      3.14159265358979323846f

typedef __attribute__((ext_vector_type(2))) float v2f;
typedef __attribute__((ext_vector_type(8))) float v8f;

#define WAVES_PER_BLOCK 4
#define ATOMS_PER_WAVE  16
#define ATOMS_PER_BLOCK (WAVES_PER_BLOCK * ATOMS_PER_WAVE)   // 64
#define TK              128     // folded contraction dim = N_TYPES(i) * N_TYPES(j) * K_MAX
#define F_STRIDE        132     // 128 + 4 pad -> conflict-free column reads

__global__ __launch_bounds__(WAVES_PER_BLOCK * 32)
void RadialDescriptor_kernel(const int* __restrict__ types,
                             const float* __restrict__ positions,
                             const long long* __restrict__ nbrs,
                             const float* __restrict__ c_table,
                             float* __restrict__ out)
{
    // Cfull[(ti,t,k)][d] = c_table[ti][t][d][k] for d<8, 0 for d in 8..15  (B matrix, shared)
    __shared__ float sC[TK][16];
    // Per-wave A tile: F~[atom][ti*32 + t*8 + k], zero outside atom's own ti block
    __shared__ float sF[WAVES_PER_BLOCK][ATOMS_PER_WAVE][F_STRIDE];

    const int tid  = threadIdx.x;
    const int lane = tid & 31;
    const int wave = tid >> 5;

    // ---- Build folded coefficient matrix Cfull (once per block) ----
    for (int e = tid; e < TK * 16; e += blockDim.x) {
        const int row = e >> 4;          // (ti,t,k) folded index
        const int n   = e & 15;          // descriptor column
        const int ti  = row >> 5;
        const int t   = (row >> 3) & 3;
        const int k   = row & 7;
        float v = 0.0f;
        if (n < N_DESC)
            v = c_table[(((ti * N_TYPES) + t) * N_DESC + n) * K_MAX + k];
        sC[row][n] = v;
    }
    // ---- Zero this wave's F~ tile ----
    for (int e = lane; e < ATOMS_PER_WAVE * F_STRIDE; e += 32)
        ((float*)sF[wave])[e] = 0.0f;
    __syncthreads();

    const int atom_base = blockIdx.x * ATOMS_PER_BLOCK + wave * ATOMS_PER_WAVE;

    // ---- Phase 1: one atom per iteration, lane = neighbor (wave32 == NN_RADIAL) ----
    for (int a = 0; a < ATOMS_PER_WAVE; ++a) {
        const int i = atom_base + a;
        if (i >= N_ATOMS) break;                      // wave-uniform
        const int   ti = types[i];
        const float xi = positions[3 * i + 0];
        const float yi = positions[3 * i + 1];
        const float zi = positions[3 * i + 2];

        const long long jraw = nbrs[(long long)i * NN + lane];
        const bool mask = (jraw != -1LL);
        int jj = mask ? (int)jraw : 0;
        jj = jj < 0 ? 0 : (jj >= N_ATOMS ? N_ATOMS - 1 : jj);

        const float dx = positions[3 * jj + 0] - xi;
        const float dy = positions[3 * jj + 1] - yi;
        const float dz = positions[3 * jj + 2] - zi;
        const float r  = sqrtf(fmaxf(dx * dx + dy * dy + dz * dz, 1e-12f));

        float fc = 0.0f;
        if (r < R_C) fc = 0.5f * cosf(PI_F * (r / R_C)) + 0.5f;
        const float hfc = mask ? 0.5f * fc : 0.0f;

        const float u = r / R_C - 1.0f;
        const float x = 2.0f * u * u - 1.0f;

        const int tj = types[jj];
        float* dst = &sF[wave][a][ti * 32 + tj * 8];

        // Chebyshev recurrence, accumulate (T_k + 1) * 0.5 * fc via LDS float atomics
        float tm2 = 1.0f, tm1 = x;
        atomicAdd(&dst[0], 2.0f * hfc);           // (1 + 1) * hfc
        atomicAdd(&dst[1], (x + 1.0f) * hfc);
        for (int k = 2; k < K_MAX; ++k) {
            const float tcur = 2.0f * x * tm1 - tm2;
            tm2 = tm1; tm1 = tcur;
            atomicAdd(&dst[k], (tcur + 1.0f) * hfc);
        }
    }
    __syncthreads();

    // ---- Phase 2: D(16x16) = F~(16x128) x Cfull(128x16) via 32 chained f32 WMMAs ----
    // A 16x4 layout: lanes 0-15 M=lane K={0,1}; lanes 16-31 M=lane-16 K={2,3}
    // B 4x16 layout: VGPR0 lanes 0-15 K=0 N=lane, lanes 16-31 K=2; VGPR1 K=1/K=3
    const int mrow = lane & 15;
    const int koff = (lane >> 4) << 1;            // 0 for lanes 0-15, 2 for 16-31

    v8f acc = {0.f, 0.f, 0.f, 0.f, 0.f, 0.f, 0.f, 0.f};
    for (int kk = 0; kk < TK / 4; ++kk) {
        const int kb = kk * 4 + koff;
        v2f A, B;
        A.x = sF[wave][mrow][kb];
        A.y = sF[wave][mrow][kb + 1];
        B.x = sC[kb][mrow];
        B.y = sC[kb + 1][mrow];
        acc = __builtin_amdgcn_wmma_f32_16x16x4_f32(
            /*neg_a=*/false, A, /*neg_b=*/false, B,
            /*c_mod=*/(short)0, acc, /*reuse_a=*/false, /*reuse_b=*/false);
    }

    // ---- Store: D element (M,N) -> VGPR M%8, lane = N + 16*(M/8) ----
    const int d = lane & 15;
    if (d < N_DESC) {
        const int mhalf = (lane >> 4) * 8;
        #pragma unroll
        for (int v = 0; v < 8; ++v) {
            const int atom = atom_base + mhalf + v;
            if (atom < N_ATOMS)
                out[atom * N_DESC + d] = acc[v];
        }
    }
}

extern "C" void kernel_launch(void* const* d_in, const int* in_sizes, int n_in,
                              void* d_out, int out_size, void* d_ws, size_t ws_size,
                              hipStream_t stream) {
    const int*       types = (const int*)d_in[0];
    const float*     pos   = (const float*)d_in[1];
    const long long* nbrs  = (const long long*)d_in[2];
    const float*     ctab  = (const float*)d_in[3];
    float*           out   = (float*)d_out;

    const int grid = (N_ATOMS + ATOMS_PER_BLOCK - 1) / ATOMS_PER_BLOCK;  // 782
    RadialDescriptor_kernel<<<grid, WAVES_PER_BLOCK * 32, 0, stream>>>(
        types, pos, nbrs, ctab, out);
}